// ChannelAttention_1271310319753
// MI455X (gfx1250) — compile-verified
//
#include <hip/hip_runtime.h>

typedef __attribute__((ext_vector_type(2))) float v2f;
typedef __attribute__((ext_vector_type(8))) float v8f;

#define SE_EPS 1e-3f
constexpr int BATCH = 64;
constexpr int HW    = 56 * 56;   // 3136
constexpr int C     = 256;
constexpr int CR    = 16;        // bottleneck
constexpr int C4    = C / 4;     // 64 float4 per channel row

// ---------------------------------------------------------------------------
// Kernel 1: partial spatial sums.  grid = BATCH * nchunk blocks, 256 threads.
// Thread layout: c4 = tid & 63 (which float4 of the 256 channels),
//                slot = tid >> 6 (4 hw rows in flight per iteration).
// Each iteration the block streams 4 contiguous rows = 4 KB, fully coalesced.
// ---------------------------------------------------------------------------
__global__ __launch_bounds__(256) void se_pool_partial(
    const float4* __restrict__ x4, float4* __restrict__ part,
    int nchunk, int rows)
{
    const int blk   = blockIdx.x;
    const int b     = blk / nchunk;
    const int chunk = blk - b * nchunk;
    const int c4    = threadIdx.x & 63;
    const int slot  = threadIdx.x >> 6;

    const size_t base = ((size_t)b * HW + (size_t)chunk * rows) * C4 + c4;

    float4 acc = make_float4(0.f, 0.f, 0.f, 0.f);
    #pragma unroll 8
    for (int r = slot; r < rows; r += 4) {
        const float4 v = x4[base + (size_t)r * C4];
        acc.x += v.x; acc.y += v.y; acc.z += v.z; acc.w += v.w;
    }

    __shared__ float4 red[256];
    red[threadIdx.x] = acc;
    __syncthreads();

    if (slot == 0) {
        const float4 a0 = red[c4];
        const float4 a1 = red[64  + c4];
        const float4 a2 = red[128 + c4];
        const float4 a3 = red[192 + c4];
        float4 o;
        o.x = (a0.x + a1.x) + (a2.x + a3.x);
        o.y = (a0.y + a1.y) + (a2.y + a3.y);
        o.z = (a0.z + a1.z) + (a2.z + a3.z);
        o.w = (a0.w + a1.w) + (a2.w + a3.w);
        part[(size_t)blk * C4 + c4] = o;
    }
}

// ---------------------------------------------------------------------------
// Kernel 2: fold chunk partials, scale by 1/HW -> s[64,256]
// ---------------------------------------------------------------------------
__global__ __launch_bounds__(256) void se_pool_finish(
    const float* __restrict__ part, float* __restrict__ s, int nchunk)
{
    const int b = blockIdx.x;
    const int c = threadIdx.x;
    const float* p = part + ((size_t)b * nchunk) * C + c;
    float acc = 0.f;
    for (int k = 0; k < nchunk; ++k) acc += p[(size_t)k * C];
    s[b * C + c] = acc * (1.0f / (float)HW);
}

// ---------------------------------------------------------------------------
// Kernel 3: excite MLP with fp32 WMMA (V_WMMA_F32_16X16X4_F32).
// 128 threads = 4 wave32; wave w owns M-tile rows [16w, 16w+16).
//
// Fragment layouts (ISA 7.12.2, 32-bit, wave32):
//   A (16x4, MxK): lane<16 -> (M=lane, K=k+0 / k+1); lane>=16 -> (K=k+2 / k+3)
//   B (4x16, KxN): lane<16 -> (K=k+0 / k+1, N=lane); lane>=16 -> (K=k+2 / k+3)
//   C/D (16x16):   vgpr r  -> M = r + 8*(lane>=16), N = lane & 15
// ---------------------------------------------------------------------------
__global__ __launch_bounds__(128) void se_mlp_wmma(
    const float* __restrict__ s,    // [64,256]
    const float* __restrict__ w1,   // [256,16]
    const float* __restrict__ g1, const float* __restrict__ bt1,
    const float* __restrict__ mn1, const float* __restrict__ vr1,
    const float* __restrict__ w2,   // [16,256]
    const float* __restrict__ g2, const float* __restrict__ bt2,
    const float* __restrict__ mn2, const float* __restrict__ vr2,
    float* __restrict__ out)        // [64,256]
{
    __shared__ float hsm[BATCH * CR];   // h[64,16]

    const int tid  = threadIdx.x;
    const int lane = tid & 31;
    const int wv   = tid >> 5;      // M-tile id (0..3)
    const int mb   = wv * 16;
    const int l    = lane & 15;
    const int hi   = lane >> 4;     // half-wave select

    // ---- GEMM1: s[64,256] @ w1[256,16] -> 16x16 tile per wave -------------
    v8f acc = {};
    for (int k0 = 0; k0 < C; k0 += 4) {
        const int ka = k0 + 2 * hi;
        v2f a, b;
        const float* srow = s + (size_t)(mb + l) * C + ka;
        a[0] = srow[0];
        a[1] = srow[1];
        b[0] = w1[ka * CR + l];
        b[1] = w1[(ka + 1) * CR + l];
        acc = __builtin_amdgcn_wmma_f32_16x16x4_f32(
            false, a, false, b, (short)0, acc, false, false);
    }

    // ---- BN1 + ReLU (column n = l), stash h in LDS -------------------------
    {
        const float sc = g1[l] * rsqrtf(vr1[l] + SE_EPS);
        const float mu = mn1[l];
        const float be = bt1[l];
        #pragma unroll
        for (int r = 0; r < 8; ++r) {
            float hv = (acc[r] - mu) * sc + be;
            hv = fmaxf(hv, 0.f);
            hsm[(mb + r + 8 * hi) * CR + l] = hv;
        }
    }
    __syncthreads();

    // ---- GEMM2: h[64,16] @ w2[16,256]; 16 N-tiles per wave -----------------
    for (int t = 0; t < 16; ++t) {
        v8f acc2 = {};
        #pragma unroll
        for (int j = 0; j < 4; ++j) {
            const int kb = 4 * j + 2 * hi;
            v2f a, b;
            a[0] = hsm[(mb + l) * CR + kb];
            a[1] = hsm[(mb + l) * CR + kb + 1];
            b[0] = w2[kb * C + t * 16 + l];
            b[1] = w2[(kb + 1) * C + t * 16 + l];
            acc2 = __builtin_amdgcn_wmma_f32_16x16x4_f32(
                false, a, false, b, (short)0, acc2, false, false);
        }
        // ---- BN2 + store (column n) ----------------------------------------
        const int n = t * 16 + l;
        const float sc = g2[n] * rsqrtf(vr2[n] + SE_EPS);
        const float mu = mn2[n];
        const float be = bt2[n];
        #pragma unroll
        for (int r = 0; r < 8; ++r) {
            out[(size_t)(mb + r + 8 * hi) * C + n] = (acc2[r] - mu) * sc + be;
        }
    }
}

// ---------------------------------------------------------------------------
extern "C" void kernel_launch(void* const* d_in, const int* in_sizes, int n_in,
                              void* d_out, int out_size, void* d_ws, size_t ws_size,
                              hipStream_t stream)
{
    const float* x   = (const float*)d_in[0];
    const float* w1  = (const float*)d_in[1];
    const float* g1  = (const float*)d_in[2];
    const float* bt1 = (const float*)d_in[3];
    const float* mn1 = (const float*)d_in[4];
    const float* vr1 = (const float*)d_in[5];
    const float* w2  = (const float*)d_in[6];
    const float* g2  = (const float*)d_in[7];
    const float* bt2 = (const float*)d_in[8];
    const float* mn2 = (const float*)d_in[9];
    const float* vr2 = (const float*)d_in[10];
    float* out = (float*)d_out;

    // pick chunk count so partials + s fit in workspace (3136 = 49*64 = 7*448)
    int nchunk = 49;
    auto need = [](int nc) {
        return (size_t)BATCH * nc * C * sizeof(float) +
               (size_t)BATCH * C * sizeof(float);
    };
    if (ws_size < need(nchunk)) nchunk = 7;
    if (ws_size < need(nchunk)) nchunk = 1;
    const int rows = HW / nchunk;

    float* part = (float*)d_ws;                          // [B*nchunk, 256]
    float* s    = part + (size_t)BATCH * nchunk * C;     // [64, 256]

    se_pool_partial<<<BATCH * nchunk, 256, 0, stream>>>(
        (const float4*)x, (float4*)part, nchunk, rows);
    se_pool_finish<<<BATCH, 256, 0, stream>>>(part, s, nchunk);
    se_mlp_wmma<<<1, 128, 0, stream>>>(
        s, w1, g1, bt1, mn1, vr1, w2, g2, bt2, mn2, vr2, out);
}